// EncoderLSTM_1314259992937
// MI455X (gfx1250) — compile-verified
//
#include <hip/hip_runtime.h>
#include <hip/hip_bf16.h>

typedef __attribute__((ext_vector_type(2))) float v2f;
typedef __attribute__((ext_vector_type(4))) float v4f;
typedef __attribute__((ext_vector_type(8))) float v8f;

#define EMBED 1024
#define ROWS_PER_BLOCK 16
#define WAVES_PER_BLOCK 8
#define K_PER_WAVE 256   // (2 * EMBED) / WAVES_PER_BLOCK

// Kernel 1: gates[4096] = W_ih @ x + b_ih + W_hh @ h + b_hh
// One block per 16 gate rows. 8 waves split the 2048-deep reduction
// (waves 0-3: W_ih vs x ; waves 4-7: W_hh vs h). Each 8-K step uses one
// B128 load of W, one B128 load of the vector, and two
// V_WMMA_F32_16X16X4_F32 accumulations (k-slots permuted consistently
// between A and B so per-lane data is contiguous).
__global__ __launch_bounds__(256) void lstm_gates_gemv_wmma(
    const int*   __restrict__ token,
    const float* __restrict__ h_state,
    const float* __restrict__ embed,
    const float* __restrict__ W_ih,
    const float* __restrict__ W_hh,
    const float* __restrict__ b_ih,
    const float* __restrict__ b_hh,
    float*       __restrict__ gates)
{
    __shared__ float part[WAVES_PER_BLOCK][ROWS_PER_BLOCK];

    const int tid  = threadIdx.x;
    const int lane = tid & 31;
    const int wave = tid >> 5;
    const int n    = lane & 15;   // B operand: N = lane % 16 (selects W row in tile)
    const int half = lane >> 4;   // lane half: supplies K-slots {0,1} or {2,3}
    const int row0 = blockIdx.x * ROWS_PER_BLOCK;

    const float* x   = embed + (size_t)token[0] * EMBED;   // embedding row
    const float* Wm  = (wave < 4) ? W_ih : W_hh;
    const float* vec = (wave < 4) ? x    : h_state;
    const int kbase  = (wave & 3) * K_PER_WAVE;

    // Per-lane contiguous B128 streams (16B aligned: kbase%8==0, +4*half).
    const float* wptr = Wm + (size_t)(row0 + n) * EMBED + kbase + 4 * half;
    const float* vptr = vec + kbase + 4 * half;

    v8f acc = {};
    #pragma unroll 4
    for (int it = 0; it < K_PER_WAVE / 8; ++it) {
        v4f a = *(const v4f*)(vptr + it * 8);   // vector chunk (broadcast A)
        v4f b = *(const v4f*)(wptr + it * 8);   // 16-row W tile (B)
        // WMMA0: actual k = {k0,k0+1 | k0+4,k0+5}; WMMA1: {k0+2,k0+3 | k0+6,k0+7}
        v2f a0 = {a.x, a.y}, a1 = {a.z, a.w};
        v2f b0 = {b.x, b.y}, b1 = {b.z, b.w};
        acc = __builtin_amdgcn_wmma_f32_16x16x4_f32(
            false, a0, false, b0, (short)0, acc, false, false);
        acc = __builtin_amdgcn_wmma_f32_16x16x4_f32(
            false, a1, false, b1, (short)0, acc, false, false);
    }

    // Column n==0 of D carries the true partials: lane 0 has rows 0-7
    // (VGPR j -> M=j), lane 16 has rows 8-15 (VGPR j -> M=8+j).
    if (n == 0) {
        #pragma unroll
        for (int j = 0; j < 8; ++j)
            part[wave][(half << 3) + j] = acc[j];
    }
    __syncthreads();

    if (tid < ROWS_PER_BLOCK) {
        float s = 0.0f;
        #pragma unroll
        for (int w = 0; w < WAVES_PER_BLOCK; ++w) s += part[w][tid];
        const int r = row0 + tid;
        gates[r] = s + b_ih[r] + b_hh[r];
    }
}

// Kernel 2: gate nonlinearity + state update (PyTorch order i, f, g, o).
__global__ __launch_bounds__(256) void lstm_pointwise(
    const float* __restrict__ gates,
    const float* __restrict__ c_state,
    float*       __restrict__ out)   // [0..1023]=h_new, [1024..2047]=c_new
{
    const int j = blockIdx.x * blockDim.x + threadIdx.x;
    if (j >= EMBED) return;
    const float gi = gates[j];
    const float gf = gates[j +     EMBED];
    const float gg = gates[j + 2 * EMBED];
    const float go = gates[j + 3 * EMBED];

    const float i = 1.0f / (1.0f + expf(-gi));
    const float f = 1.0f / (1.0f + expf(-gf));
    const float g = tanhf(gg);
    const float o = 1.0f / (1.0f + expf(-go));

    const float c_new = f * c_state[j] + i * g;
    const float h_new = o * tanhf(c_new);

    out[j]         = h_new;
    out[EMBED + j] = c_new;
}

extern "C" void kernel_launch(void* const* d_in, const int* in_sizes, int n_in,
                              void* d_out, int out_size, void* d_ws, size_t ws_size,
                              hipStream_t stream) {
    // setup_inputs order:
    // 0: input (int token), 1: hidden_state [1,1,1024], 2: cell_state [1,1,1024],
    // 3: embed_table [50257,1024], 4: W_ih [4096,1024], 5: W_hh [4096,1024],
    // 6: b_ih [4096], 7: b_hh [4096]
    const int*   token   = (const int*)  d_in[0];
    const float* h_state = (const float*)d_in[1];
    const float* c_state = (const float*)d_in[2];
    const float* embed   = (const float*)d_in[3];
    const float* W_ih    = (const float*)d_in[4];
    const float* W_hh    = (const float*)d_in[5];
    const float* b_ih    = (const float*)d_in[6];
    const float* b_hh    = (const float*)d_in[7];

    float* gates = (float*)d_ws;          // 4096 floats = 16 KB scratch
    float* out   = (float*)d_out;         // 2048 floats: h_new ++ c_new

    lstm_gates_gemv_wmma<<<4 * EMBED / ROWS_PER_BLOCK, 256, 0, stream>>>(
        token, h_state, embed, W_ih, W_hh, b_ih, b_hh, gates);

    lstm_pointwise<<<EMBED / 256, 256, 0, stream>>>(gates, c_state, out);
}